// RGCN_76501957476384
// MI455X (gfx1250) — compile-verified
//
#include <hip/hip_runtime.h>

// RGCN on MI455X (gfx1250). Bandwidth-bound on edge scatter/gather
// (~10 GB total @ 23.3 TB/s); dense transforms done with native fp32
// WMMA (V_WMMA_F32_16X16X4_F32) to keep reference fp32 numerics.
// LDS weight tiles are stored pair-interleaved so each WMMA B fragment
// is one ds_load_b64 straight into the operand register pair.

#define R_ 8
#define N_ 50000
#define E_ 400000
#define D_ 128
#define C_ 16
#define B_ 4

typedef float v2f __attribute__((ext_vector_type(2)));
typedef float v8f __attribute__((ext_vector_type(8)));

static inline int cdiv_i(int a, int b) { return (a + b - 1) / b; }

// ---------------- utility kernels ----------------

__global__ void zero_kernel(float* __restrict__ p, int total) {
  int i = blockIdx.x * blockDim.x + threadIdx.x;
  if (i < total) p[i] = 0.0f;
}

__global__ void deg_count_kernel(float* __restrict__ deg,           // [R*N]
                                 const int* __restrict__ dstAll,    // [R*E]
                                 int total, int Eper, int Nn) {
  int i = blockIdx.x * blockDim.x + threadIdx.x;
  if (i >= total) return;
  int r = i / Eper;
  atomicAdd(deg + (size_t)r * Nn + dstAll[i], 1.0f);
}

__global__ void inv_deg_kernel(float* __restrict__ deg, int total) {
  int i = blockIdx.x * blockDim.x + threadIdx.x;
  if (i >= total) return;
  deg[i] = 1.0f / fmaxf(deg[i], 1.0f);
}

// outW[r][j] = sum_b comp[r][b] * basis[b][j],  j in [0, M)
__global__ void combine_kernel(float* __restrict__ outW,
                               const float* __restrict__ basis,
                               const float* __restrict__ comp,
                               int Bn, int M, int total) {
  int i = blockIdx.x * blockDim.x + threadIdx.x;
  if (i >= total) return;
  int r = i / M, j = i - r * M;
  float s = 0.0f;
#pragma unroll
  for (int b = 0; b < B_; ++b) s += comp[r * Bn + b] * basis[(size_t)b * M + j];
  outW[i] = s;
}

__global__ void bias_act_kernel(float* __restrict__ x, const float* __restrict__ bias,
                                int Dp, int total, int doRelu) {
  int i = blockIdx.x * blockDim.x + threadIdx.x;
  if (i >= total) return;
  float v = x[i] + bias[i % Dp];
  x[i] = doRelu ? fmaxf(v, 0.0f) : v;
}

// copy_u + mean aggregation for one relation, accumulated into outAcc.
// One thread per (edge, 4-float chunk); message = feat[src]*inv[dst].
__global__ void scatter_mean_kernel(float* __restrict__ outAcc,      // [N, 4*chunks]
                                    const float* __restrict__ feat,  // [N, 4*chunks]
                                    const int* __restrict__ src,
                                    const int* __restrict__ dst,
                                    const float* __restrict__ inv,   // [N]
                                    int nE, int chunks) {
  int i = blockIdx.x * blockDim.x + threadIdx.x;
  int total = nE * chunks;
  if (i >= total) return;
  int e = i / chunks, c = i - e * chunks;
  int s = src[e], d = dst[e];
  float w = inv[d];
  float4 v = ((const float4*)feat)[(size_t)s * chunks + c];
  float* o = outAcc + ((size_t)d * chunks + c) * 4;
  atomicAdd(o + 0, v.x * w);
  atomicAdd(o + 1, v.y * w);
  atomicAdd(o + 2, v.z * w);
  atomicAdd(o + 3, v.w * w);
}

// ---------------- WMMA GEMM kernels (fp32, 16x16x4) ----------------
// LDS layout (pair-interleaved): Wlds[j*2*COLS + 2*c + s] = W[(2j+s)*COLS + c]
// so the B fragment for k-step k, half-lane h (j0 = k/2 + h) at column c is
// the contiguous pair Wlds[j0*2*COLS + 2*c .. +1] -> one ds_load_b64.

// Xout[tile*16 .. +15][0..127] = Hin[rows][0..127] @ W[128x128]
// 8 waves/block, one 16-row tile per wave, whole W cached in LDS (64 KB).
__global__ void __launch_bounds__(256)
rgcn_gemm_d128(const float* __restrict__ Hin, const float* __restrict__ W,
               float* __restrict__ Xout, int nRowTiles) {
  __shared__ float Wlds[D_ * D_];  // 64 KB, pair-interleaved
  for (int idx = threadIdx.x; idx < 64 * 64; idx += 256) {
    int j = idx >> 6, cp = idx & 63;   // j = k/2, cp = c/2
    int c = cp * 2;
    float2 r0 = *(const float2*)(W + (size_t)(2 * j) * D_ + c);
    float2 r1 = *(const float2*)(W + (size_t)(2 * j + 1) * D_ + c);
    *(float4*)(Wlds + j * 256 + 2 * c) = make_float4(r0.x, r1.x, r0.y, r1.y);
  }
  __syncthreads();

  int wave = threadIdx.x >> 5;
  int lane = threadIdx.x & 31;
  int tile = blockIdx.x * 8 + wave;        // wave-uniform -> EXEC all-1s for WMMA
  if (tile >= nRowTiles) return;
  int half = lane >> 4;                    // 0: K,K+1   1: K+2,K+3
  int m = lane & 15;

  const float* arow = Hin + (size_t)(tile * 16 + m) * D_ + 2 * half;
  const float* bcol = Wlds + 2 * m + half * 256;   // half folds into j0
  v8f acc[8] = {};
  for (int k = 0; k < D_; k += 4) {
    float2 av = *(const float2*)(arow + k);        // A frag: [M=m][k+2h, k+1+2h]
    v2f a; a[0] = av.x; a[1] = av.y;
    const float* bb = bcol + (k >> 1) * 256;       // j0 = k/2 + half
#pragma unroll
    for (int t = 0; t < 8; ++t) {                  // 8 column tiles of 16
      v2f b = *(const v2f*)(bb + t * 32);          // (W[k+2h][c], W[k+1+2h][c])
      acc[t] = __builtin_amdgcn_wmma_f32_16x16x4_f32(
          false, a, false, b, (short)0, acc[t], false, false);
    }
  }
  // C/D layout: VGPR v -> row tile*16 + v + 8*half, col t*16 + m
  float* orow = Xout + (size_t)(tile * 16 + 8 * half) * D_ + m;
#pragma unroll
  for (int t = 0; t < 8; ++t)
#pragma unroll
    for (int v = 0; v < 8; ++v)
      orow[(size_t)v * D_ + t * 16] = acc[t][v];
}

// Xout[tile*16 .. +15][0..15] = Hin[rows][0..127] @ W[128x16]
__global__ void __launch_bounds__(256)
rgcn_gemm_d16(const float* __restrict__ Hin, const float* __restrict__ W,
              float* __restrict__ Xout, int nRowTiles) {
  __shared__ float Wlds[D_ * C_];  // 8 KB, pair-interleaved, conflict-free
  for (int idx = threadIdx.x; idx < 64 * 8; idx += 256) {
    int j = idx >> 3, cp = idx & 7;
    int c = cp * 2;
    float2 r0 = *(const float2*)(W + (size_t)(2 * j) * C_ + c);
    float2 r1 = *(const float2*)(W + (size_t)(2 * j + 1) * C_ + c);
    *(float4*)(Wlds + j * 32 + 2 * c) = make_float4(r0.x, r1.x, r0.y, r1.y);
  }
  __syncthreads();

  int wave = threadIdx.x >> 5;
  int lane = threadIdx.x & 31;
  int tile = blockIdx.x * 8 + wave;
  if (tile >= nRowTiles) return;
  int half = lane >> 4;
  int m = lane & 15;

  const float* arow = Hin + (size_t)(tile * 16 + m) * D_ + 2 * half;
  const float* bcol = Wlds + 2 * m + half * 32;
  v8f acc = {};
  for (int k = 0; k < D_; k += 4) {
    float2 av = *(const float2*)(arow + k);
    v2f a; a[0] = av.x; a[1] = av.y;
    v2f b = *(const v2f*)(bcol + (k >> 1) * 32);
    acc = __builtin_amdgcn_wmma_f32_16x16x4_f32(
        false, a, false, b, (short)0, acc, false, false);
  }
  float* orow = Xout + (size_t)(tile * 16 + 8 * half) * C_ + m;
#pragma unroll
  for (int v = 0; v < 8; ++v) orow[v * C_] = acc[v];
}

// ---------------- orchestration ----------------

extern "C" void kernel_launch(void* const* d_in, const int* in_sizes, int n_in,
                              void* d_out, int out_size, void* d_ws, size_t ws_size,
                              hipStream_t stream) {
  (void)in_sizes; (void)n_in; (void)out_size; (void)ws_size;
  const int*   edge_src = (const int*)d_in[0];    // [R,E]
  const int*   edge_dst = (const int*)d_in[1];    // [R,E]
  const float* embeds   = (const float*)d_in[2];  // [R,N,D]
  const float* emb_bias = (const float*)d_in[3];  // [D]
  const float* w1_basis = (const float*)d_in[4];  // [B,D,D]
  const float* w1_comp  = (const float*)d_in[5];  // [R,B]
  const float* b1       = (const float*)d_in[6];  // [D]
  const float* w2_basis = (const float*)d_in[7];  // [B,D,C]
  const float* w2_comp  = (const float*)d_in[8];  // [R,B]
  const float* b2       = (const float*)d_in[9];  // [C]
  float* out = (float*)d_out;                     // [N,C]

  float* ws  = (float*)d_ws;
  float* inv = ws;                          // R*N
  float* h   = inv + (size_t)R_ * N_;       // N*D
  float* h1  = h   + (size_t)N_ * D_;       // N*D
  float* xr  = h1  + (size_t)N_ * D_;       // N*D
  float* xr2 = xr  + (size_t)N_ * D_;       // N*C
  float* W1  = xr2 + (size_t)N_ * C_;       // R*D*D
  float* W2  = W1  + (size_t)R_ * D_ * D_;  // R*D*C

  const int TPB = 256;
  const int rowTiles = N_ / 16;             // 3125 exact tiles, no tail
  const int gemmBlocks = cdiv_i(rowTiles, 8);

  // degree -> inverse degree
  zero_kernel<<<cdiv_i(R_ * N_, TPB), TPB, 0, stream>>>(inv, R_ * N_);
  deg_count_kernel<<<cdiv_i(R_ * E_, TPB), TPB, 0, stream>>>(inv, edge_dst, R_ * E_, E_, N_);
  inv_deg_kernel<<<cdiv_i(R_ * N_, TPB), TPB, 0, stream>>>(inv, R_ * N_);

  // basis-combined relation weights
  combine_kernel<<<cdiv_i(R_ * D_ * D_, TPB), TPB, 0, stream>>>(
      W1, w1_basis, w1_comp, B_, D_ * D_, R_ * D_ * D_);
  combine_kernel<<<cdiv_i(R_ * D_ * C_, TPB), TPB, 0, stream>>>(
      W2, w2_basis, w2_comp, B_, D_ * C_, R_ * D_ * C_);

  // ---- embed layer: sum_r mean_agg(embeds[r]) ; relu(+emb_bias) ----
  zero_kernel<<<cdiv_i(N_ * D_, TPB), TPB, 0, stream>>>(h, N_ * D_);
  for (int r = 0; r < R_; ++r) {
    scatter_mean_kernel<<<cdiv_i(E_ * (D_ / 4), TPB), TPB, 0, stream>>>(
        h, embeds + (size_t)r * N_ * D_, edge_src + (size_t)r * E_,
        edge_dst + (size_t)r * E_, inv + (size_t)r * N_, E_, D_ / 4);
  }
  bias_act_kernel<<<cdiv_i(N_ * D_, TPB), TPB, 0, stream>>>(h, emb_bias, D_, N_ * D_, 1);

  // ---- hidden layer: sum_r mean_agg(h @ W1[r]) ; relu(+b1) ----
  zero_kernel<<<cdiv_i(N_ * D_, TPB), TPB, 0, stream>>>(h1, N_ * D_);
  for (int r = 0; r < R_; ++r) {
    rgcn_gemm_d128<<<gemmBlocks, TPB, 0, stream>>>(h, W1 + (size_t)r * D_ * D_, xr, rowTiles);
    scatter_mean_kernel<<<cdiv_i(E_ * (D_ / 4), TPB), TPB, 0, stream>>>(
        h1, xr, edge_src + (size_t)r * E_, edge_dst + (size_t)r * E_,
        inv + (size_t)r * N_, E_, D_ / 4);
  }
  bias_act_kernel<<<cdiv_i(N_ * D_, TPB), TPB, 0, stream>>>(h1, b1, D_, N_ * D_, 1);

  // ---- output layer: sum_r mean_agg(h1 @ W2[r]) ; + b2 ----
  zero_kernel<<<cdiv_i(N_ * C_, TPB), TPB, 0, stream>>>(out, N_ * C_);
  for (int r = 0; r < R_; ++r) {
    rgcn_gemm_d16<<<gemmBlocks, TPB, 0, stream>>>(h1, W2 + (size_t)r * D_ * C_, xr2, rowTiles);
    scatter_mean_kernel<<<cdiv_i(E_ * (C_ / 4), TPB), TPB, 0, stream>>>(
        out, xr2, edge_src + (size_t)r * E_, edge_dst + (size_t)r * E_,
        inv + (size_t)r * N_, E_, C_ / 4);
  }
  bias_act_kernel<<<cdiv_i(N_ * C_, TPB), TPB, 0, stream>>>(out, b2, C_, N_ * C_, 0);
}